// heatsolve_67929202753587
// MI455X (gfx1250) — compile-verified
//
#include <hip/hip_runtime.h>
#include <cstdint>

// Problem constants from the reference (fixed by setup_inputs).
#define NX      1024
#define TSTEPS  256
#define CORE    64                  // output tile core per block
#define HALO    8                   // temporal-blocking halo == max fused steps
#define BDIM    (CORE + 2 * HALO)   // 80: LDS buffer dim
#define BCH     (BDIM / 4)          // 20 float4 chunks per buffer row
#define NTILE   (NX / CORE)         // 16 tiles per dimension
#define NBLK    (NTILE * NTILE)     // 256 blocks per launch
#define NTHR    256                 // 8 wave32 per block

// ---------------------------------------------------------------------------
// Slice 0: u_init = u0 with all four boundaries zeroed. One float4/thread.
// ---------------------------------------------------------------------------
__global__ __launch_bounds__(NTHR) void heat_init_kernel(
    const float* __restrict__ u0, float* __restrict__ out0) {
  const int v4  = blockIdx.x * blockDim.x + threadIdx.x;
  const int idx = v4 * 4;
  const int y   = idx >> 10;
  const int x   = idx & (NX - 1);
  float4 v = reinterpret_cast<const float4*>(u0)[v4];
  if (y == 0 || y == NX - 1) {
    v = make_float4(0.f, 0.f, 0.f, 0.f);
  } else {
    if (x == 0)          v.x = 0.f;
    if (x + 3 == NX - 1) v.w = 0.f;
  }
  reinterpret_cast<float4*>(out0)[v4] = v;
}

// ---------------------------------------------------------------------------
// Fused temporal block: read slice tbase, perform nsteps (<= HALO) Euler steps
// entirely in LDS (80x80 redundant-halo tile, ping-pong buffers), writing the
// exact 64x64 core of each intermediate slice tbase+1..tbase+nsteps.
// Staging uses gfx1250 async global->LDS b128 copies (ASYNCcnt).
// ---------------------------------------------------------------------------
__global__ __launch_bounds__(NTHR) void heat_fused_kernel(
    float* out, int tbase, int nsteps,
    const float* __restrict__ alpha_p, const int* __restrict__ T_p) {
  __shared__ float bufA[BDIM][BDIM];
  __shared__ float bufB[BDIM][BDIM];

  const int tid = threadIdx.x;
  const int bx  = blockIdx.x % NTILE;
  const int by  = blockIdx.x / NTILE;
  const int gx0 = bx * CORE;   // core origin (global coords)
  const int gy0 = by * CORE;

  // gamma = alpha * dt / dx^2 (alpha, T are device-resident scalars)
  const float alpha = alpha_p[0];
  const float Tval  = (float)T_p[0];
  const float dx    = 6.0f / (float)(NX - 1);
  const float dt    = Tval / (float)(TSTEPS - 1);
  const float gamma = alpha * dt / (dx * dx);

  const float* uin = out + (size_t)tbase * NX * NX;

  // --- async stage: BDIM x BDIM tile (address-clamped) -> bufA -------------
  for (int i = tid; i < BDIM * BCH; i += NTHR) {
    const int ly = i / BCH, c = i % BCH;
    int gy = gy0 - HALO + ly;
    gy = gy < 0 ? 0 : (gy > NX - 1 ? NX - 1 : gy);
    int gx = gx0 - HALO + 4 * c;                 // multiple of 4
    gx = gx < 0 ? 0 : (gx > NX - 4 ? NX - 4 : gx);
    uint32_t lds_off = (uint32_t)(uintptr_t)(&bufA[ly][4 * c]);
    uint64_t gaddr   = (uint64_t)(uintptr_t)(uin + (size_t)gy * NX + gx);
    asm volatile("global_load_async_to_lds_b128 %0, %1, off"
                 :: "v"(lds_off), "v"(gaddr) : "memory");
  }
  asm volatile("s_wait_asynccnt 0" ::: "memory");
  __syncthreads();

  float (*A)[BDIM] = bufA;
  float (*B)[BDIM] = bufB;

  for (int k = 1; k <= nsteps; ++k) {
    // ---- B = step(A) over the whole buffer (outer rings go stale at 1
    //      ring/step; core stays exact since HALO >= nsteps) ----------------
    for (int i = tid; i < BDIM * BCH; i += NTHR) {
      const int ly = i / BCH, c = i % BCH;
      float4 o;
      if (ly == 0 || ly == BDIM - 1) {
        o = make_float4(0.f, 0.f, 0.f, 0.f);
      } else {
        const int xb = 4 * c;
        float4 ce = *reinterpret_cast<const float4*>(&A[ly    ][xb]);
        float4 up = *reinterpret_cast<const float4*>(&A[ly - 1][xb]);
        float4 dn = *reinterpret_cast<const float4*>(&A[ly + 1][xb]);
        const float lm = (c == 0)       ? 0.f : A[ly][xb - 1];
        const float rp = (c == BCH - 1) ? 0.f : A[ly][xb + 4];
        o.x = ce.x + gamma * (up.x + dn.x + lm   + ce.y - 4.f * ce.x);
        o.y = ce.y + gamma * (up.y + dn.y + ce.x + ce.z - 4.f * ce.y);
        o.z = ce.z + gamma * (up.z + dn.z + ce.y + ce.w - 4.f * ce.z);
        o.w = ce.w + gamma * (up.w + dn.w + ce.z + rp   - 4.f * ce.w);
        // force boundary / out-of-domain cells to zero every step
        const int gy = gy0 - HALO + ly;
        const int gx = gx0 - HALO + xb;
        if (gy <= 0 || gy >= NX - 1) {
          o = make_float4(0.f, 0.f, 0.f, 0.f);
        } else {
          if (gx     <= 0 || gx     >= NX - 1) o.x = 0.f;
          if (gx + 1 <= 0 || gx + 1 >= NX - 1) o.y = 0.f;
          if (gx + 2 <= 0 || gx + 2 >= NX - 1) o.z = 0.f;
          if (gx + 3 <= 0 || gx + 3 >= NX - 1) o.w = 0.f;
        }
      }
      *reinterpret_cast<float4*>(&B[ly][4 * c]) = o;
    }
    __syncthreads();   // B complete for all waves

    // ---- stream the exact 64x64 core of slice tbase+k to HBM -------------
    float* uslice = out + (size_t)(tbase + k) * NX * NX;
    for (int i = tid; i < CORE * (CORE / 4); i += NTHR) {
      const int cy = i / (CORE / 4), c4 = i % (CORE / 4);
      float4 v = *reinterpret_cast<const float4*>(&B[HALO + cy][HALO + 4 * c4]);
      *reinterpret_cast<float4*>(uslice + (size_t)(gy0 + cy) * NX + gx0 + 4 * c4) = v;
    }

    // ping-pong (safe with one barrier: step-k reads of A all precede the
    // barrier; the store loop touches only B; step k+1 writes old A)
    float (*tmpP)[BDIM] = A; A = B; B = tmpP;
  }
}

// ---------------------------------------------------------------------------
// inputs (setup_inputs order): u0[1024*1024] f32, alpha[1] f32, T[1] i32,
//                              nX[1] i32, time_steps[1] i32
// d_out: [256, 1024, 1024] f32
// ---------------------------------------------------------------------------
extern "C" void kernel_launch(void* const* d_in, const int* in_sizes, int n_in,
                              void* d_out, int out_size, void* d_ws, size_t ws_size,
                              hipStream_t stream) {
  (void)in_sizes; (void)n_in; (void)out_size; (void)d_ws; (void)ws_size;

  const float* u0      = (const float*)d_in[0];
  const float* alpha_p = (const float*)d_in[1];
  const int*   T_p     = (const int*)d_in[2];
  float*       out     = (float*)d_out;

  // slice 0: masked copy of u0
  heat_init_kernel<<<(NX * NX / 4) / NTHR, NTHR, 0, stream>>>(u0, out);

  // slices 1..255 in fused chunks of up to HALO steps (32 launches total)
  int t = 0;
  while (t < TSTEPS - 1) {
    const int s = (TSTEPS - 1 - t) < HALO ? (TSTEPS - 1 - t) : HALO;
    heat_fused_kernel<<<NBLK, NTHR, 0, stream>>>(out, t, s, alpha_p, T_p);
    t += s;
  }
}